// ContrastiveEnergyLearning_37950331027871
// MI455X (gfx1250) — compile-verified
//
#include <hip/hip_runtime.h>
#include <math.h>

#define D_MODEL 256
#define NNEG    16
#define NCAND   17
#define TEMP    0.07f

typedef __attribute__((ext_vector_type(2))) float v2f;
typedef __attribute__((ext_vector_type(8))) float v8f;

#define WAVES_PER_BLOCK 4
#define TILE_STRIDE 20   // 16 cols + 4 pad floats -> conflict-free 64-bank access

__device__ __forceinline__ v8f wmma_f32(v2f a, v2f b, v8f c) {
  // D = A(16x4 f32) * B(4x16 f32) + C(16x16 f32)
  return __builtin_amdgcn_wmma_f32_16x16x4_f32(false, a, false, b, (short)0, c,
                                               false, false);
}

__device__ __forceinline__ float gelu_exact(float x) {
  return 0.5f * x * (1.0f + erff(x * 0.70710678118654752f));
}

// ---------------------------------------------------------------------------
// Kernel 1: per-row MLP energy via FP32 WMMA. One wave32 handles 16 rows.
// ---------------------------------------------------------------------------
__global__ __launch_bounds__(32 * WAVES_PER_BLOCK)
void energy_mlp_kernel(const float* __restrict__ anchor,
                       const float* __restrict__ positive,
                       const float* __restrict__ negatives,
                       const float* __restrict__ W1, const float* __restrict__ b1,
                       const float* __restrict__ W2, const float* __restrict__ b2,
                       const float* __restrict__ W3, const float* __restrict__ b3,
                       const float* __restrict__ W4, const float* __restrict__ b4,
                       float* __restrict__ energies, int ntiles)
{
  __shared__ float lds_bounce[WAVES_PER_BLOCK][16 * TILE_STRIDE];

  const int lane = threadIdx.x & 31;
  const int wave = threadIdx.x >> 5;
  const int tile = blockIdx.x * WAVES_PER_BLOCK + wave;
  if (tile >= ntiles) return;              // wave-uniform: EXEC stays all-ones

  const int m  = lane & 15;                // matrix row index for this lane
  const int kp = lane >> 4;                // k-pair selector (0 or 1)
  const int row  = tile * 16 + m;          // global row in [0, B*17)
  const int bidx = row / NCAND;
  const int j    = row - bidx * NCAND;

  const float* xrow = anchor + (size_t)bidx * D_MODEL;
  const float* yrow = (j == 0)
      ? (positive + (size_t)bidx * D_MODEL)
      : (negatives + ((size_t)bidx * NNEG + (size_t)(j - 1)) * D_MODEL);

  float* tlds = &lds_bounce[wave][0];

  // ---- layer-2 accumulators (16x128 tile), bias-initialized -------------
  v8f acc2[8];
#pragma unroll
  for (int n2 = 0; n2 < 8; ++n2) {
    float bv = b2[n2 * 16 + m];
    acc2[n2] = (v8f){bv, bv, bv, bv, bv, bv, bv, bv};
  }

  const float* xbase  = xrow + 2 * kp;
  const float* ybase  = yrow + 2 * kp;
  const float* w1base = W1 + (size_t)m * 512 + 2 * kp;

  // ==== stage 1: h1 column tiles, fused into layer-2 accumulation ========
  for (int ct = 0; ct < 16; ++ct) {
    float bv1 = b1[ct * 16 + m];
    v8f acc1 = (v8f){bv1, bv1, bv1, bv1, bv1, bv1, bv1, bv1};
    const float* wp = w1base + (size_t)ct * 16 * 512;

#pragma unroll 4
    for (int k = 0; k < 256; k += 4) {          // x half of joint input
      v2f a  = *(const v2f*)(xbase + k);
      v2f bw = *(const v2f*)(wp + k);
      acc1 = wmma_f32(a, bw, acc1);
    }
#pragma unroll 4
    for (int k = 0; k < 256; k += 4) {          // y half of joint input
      v2f a  = *(const v2f*)(ybase + k);
      v2f bw = *(const v2f*)(wp + 256 + k);
      acc1 = wmma_f32(a, bw, acc1);
    }

    // GELU + C-layout -> row-major bounce through LDS
#pragma unroll
    for (int v = 0; v < 8; ++v)
      tlds[(v + 8 * kp) * TILE_STRIDE + m] = gelu_exact(acc1[v]);
    asm volatile("s_wait_dscnt 0" ::: "memory");

    // consume this 16-wide K slice into layer 2
#pragma unroll
    for (int kk = 0; kk < 4; ++kk) {
      v2f a2 = *(const v2f*)(tlds + m * TILE_STRIDE + 4 * kk + 2 * kp);
      const float* w2p = W2 + (size_t)m * 256 + (ct * 16 + 4 * kk) + 2 * kp;
#pragma unroll
      for (int n2 = 0; n2 < 8; ++n2) {
        v2f bw = *(const v2f*)(w2p + n2 * 16 * 256);
        acc2[n2] = wmma_f32(a2, bw, acc2[n2]);
      }
    }
    asm volatile("s_wait_dscnt 0" ::: "memory");
  }

  // ==== stage 2: h2 tiles fused into layer-3 accumulation ================
  v8f acc3[4];
#pragma unroll
  for (int n3 = 0; n3 < 4; ++n3) {
    float bv = b3[n3 * 16 + m];
    acc3[n3] = (v8f){bv, bv, bv, bv, bv, bv, bv, bv};
  }

  for (int ct = 0; ct < 8; ++ct) {
#pragma unroll
    for (int v = 0; v < 8; ++v)
      tlds[(v + 8 * kp) * TILE_STRIDE + m] = gelu_exact(acc2[ct][v]);
    asm volatile("s_wait_dscnt 0" ::: "memory");

#pragma unroll
    for (int kk = 0; kk < 4; ++kk) {
      v2f a2 = *(const v2f*)(tlds + m * TILE_STRIDE + 4 * kk + 2 * kp);
      const float* w3p = W3 + (size_t)m * 128 + (ct * 16 + 4 * kk) + 2 * kp;
#pragma unroll
      for (int n3 = 0; n3 < 4; ++n3) {
        v2f bw = *(const v2f*)(w3p + n3 * 16 * 128);
        acc3[n3] = wmma_f32(a2, bw, acc3[n3]);
      }
    }
    asm volatile("s_wait_dscnt 0" ::: "memory");
  }

  // ==== stage 3: h3 tiles + final 64->1 dot product ======================
  float e = 0.0f;
  for (int ct = 0; ct < 4; ++ct) {
#pragma unroll
    for (int v = 0; v < 8; ++v)
      tlds[(v + 8 * kp) * TILE_STRIDE + m] = gelu_exact(acc3[ct][v]);
    asm volatile("s_wait_dscnt 0" ::: "memory");

#pragma unroll
    for (int n = 0; n < 16; ++n)
      e += tlds[m * TILE_STRIDE + n] * W4[ct * 16 + n];
    asm volatile("s_wait_dscnt 0" ::: "memory");
  }
  e += b4[0];

  if (lane < 16) energies[row] = e;
}

// ---------------------------------------------------------------------------
// Kernel 2: per-batch-element softmax-CE stats; deterministic block reduce.
// ---------------------------------------------------------------------------
__global__ __launch_bounds__(256)
void reduce_rows_kernel(const float* __restrict__ energies,
                        float* __restrict__ partials)
{
  __shared__ float sd[256];
  const int t = threadIdx.x;
  const size_t b = (size_t)blockIdx.x * 256 + t;
  const float* e = energies + b * NCAND;

  float vals[NCAND];
#pragma unroll
  for (int jj = 0; jj < NCAND; ++jj) vals[jj] = e[jj];

  const float invT = 1.0f / TEMP;
  float lg[NCAND];
  lg[0] = -vals[0] * invT;
  float mx = lg[0];
  float negsum = 0.0f;
  float emin = vals[1];
#pragma unroll
  for (int jj = 1; jj < NCAND; ++jj) {
    negsum += vals[jj];
    emin = fminf(emin, vals[jj]);
    lg[jj] = -vals[jj] * invT;
    mx = fmaxf(mx, lg[jj]);
  }
  float s = 0.0f;
#pragma unroll
  for (int jj = 0; jj < NCAND; ++jj) s += expf(lg[jj] - mx);
  float loss_i = (mx + logf(s)) - lg[0];
  float acc_i  = (vals[0] <= emin) ? 1.0f : 0.0f;

  float q[4] = {loss_i, vals[0], negsum, acc_i};
#pragma unroll
  for (int c = 0; c < 4; ++c) {
    sd[t] = q[c];
    __syncthreads();
    for (int s2 = 128; s2 > 0; s2 >>= 1) {
      if (t < s2) sd[t] += sd[t + s2];
      __syncthreads();
    }
    if (t == 0) partials[(size_t)blockIdx.x * 4 + c] = sd[0];
    __syncthreads();
  }
}

// ---------------------------------------------------------------------------
// Kernel 3: tiny deterministic finalize.
// ---------------------------------------------------------------------------
__global__ void finalize_kernel(const float* __restrict__ partials, int nblk,
                                float* __restrict__ out, int B)
{
  if (threadIdx.x == 0 && blockIdx.x == 0) {
    double s0 = 0, s1 = 0, s2 = 0, s3 = 0;
    for (int i = 0; i < nblk; ++i) {
      s0 += partials[4 * i + 0];
      s1 += partials[4 * i + 1];
      s2 += partials[4 * i + 2];
      s3 += partials[4 * i + 3];
    }
    out[0] = (float)(s0 / (double)B);                    // loss
    out[1] = (float)(s1 / (double)B);                    // pos_energy
    out[2] = (float)(s2 / ((double)B * (double)NNEG));   // neg_energy
    out[3] = (float)(s3 / (double)B);                    // accuracy
  }
}

// ---------------------------------------------------------------------------
extern "C" void kernel_launch(void* const* d_in, const int* in_sizes, int n_in,
                              void* d_out, int out_size, void* d_ws, size_t ws_size,
                              hipStream_t stream) {
  const float* anchor    = (const float*)d_in[0];
  const float* positive  = (const float*)d_in[1];
  const float* negatives = (const float*)d_in[2];
  const float* W1 = (const float*)d_in[3];
  const float* b1 = (const float*)d_in[4];
  const float* W2 = (const float*)d_in[5];
  const float* b2 = (const float*)d_in[6];
  const float* W3 = (const float*)d_in[7];
  const float* b3 = (const float*)d_in[8];
  const float* W4 = (const float*)d_in[9];
  const float* b4 = (const float*)d_in[10];

  const int B      = in_sizes[0] / D_MODEL;   // 32768
  const int nrows  = B * NCAND;               // 557056
  const int ntiles = nrows / 16;              // 34816

  float* energies = (float*)d_ws;
  float* partials = energies + nrows;

  const int blocks1 = (ntiles + WAVES_PER_BLOCK - 1) / WAVES_PER_BLOCK;
  energy_mlp_kernel<<<blocks1, 32 * WAVES_PER_BLOCK, 0, stream>>>(
      anchor, positive, negatives, W1, b1, W2, b2, W3, b3, W4, b4,
      energies, ntiles);

  const int blocks2 = B / 256;
  reduce_rows_kernel<<<blocks2, 256, 0, stream>>>(energies, partials);
  finalize_kernel<<<1, 1, 0, stream>>>(partials, blocks2, (float*)d_out, B);
}